// GCN_Net_1Layer_75496935129210
// MI455X (gfx1250) — compile-verified
//
#include <hip/hip_runtime.h>

#define F_IN 128
#define F_OUT 32

typedef float v2f __attribute__((ext_vector_type(2)));
typedef float v8f __attribute__((ext_vector_type(8)));

// ---------------------------------------------------------------- degree ----
__global__ void gcn_init_deg(float* __restrict__ deg, int N) {
    int i = blockIdx.x * blockDim.x + threadIdx.x;
    if (i < N) deg[i] = 1.0f;                       // self loop
}

__global__ void gcn_count_deg(const int* __restrict__ dst,
                              float* __restrict__ deg, int E) {
    int e = blockIdx.x * blockDim.x + threadIdx.x;
    if (e < E) atomicAdd(&deg[dst[e]], 1.0f);
}

__global__ void gcn_dinv(const float* __restrict__ deg,
                         float* __restrict__ dinv, int N) {
    int i = blockIdx.x * blockDim.x + threadIdx.x;
    if (i < N) {
        float d = deg[i];
        dinv[i] = (d > 0.0f) ? rsqrtf(d) : 0.0f;
    }
}

// ------------------------------------------------- h = x @ W via f32 WMMA ---
// One wave computes a 16x32 tile of h. K loop: 32 steps of 16x16x4 f32 WMMA,
// two accumulators for the two 16-wide N halves. Layouts per ISA 7.12.2:
//   A (16x4 f32): lane m = lane&15, holds K = 2*(lane>>4) + {0,1}
//   B (4x16 f32): lane n = lane&15, holds K = 2*(lane>>4) + {0,1}
//   C/D (16x16):  lane n = lane&15, VGPR r -> M = r + 8*(lane>>4)
__global__ void gcn_gemm_wmma(const float* __restrict__ x,
                              const float* __restrict__ W,
                              float* __restrict__ h, int N) {
    const int lane  = threadIdx.x & 31;
    const int wave  = threadIdx.x >> 5;
    const int mtile = blockIdx.x * 8 + wave;
    const int mtiles = (N + 15) >> 4;
    if (mtile >= mtiles) return;                    // wave-uniform: EXEC stays all-1s

    const int row0 = mtile << 4;
    const int lm = lane & 15;                       // M (for A) / N (for B,C,D)
    const int lk = (lane >> 4) << 1;                // K sub-offset: 0 or 2

    int arow = row0 + lm;
    if (arow >= N) arow = N - 1;                    // clamped load; store is guarded
    const float* __restrict__ aptr = x + (size_t)arow * F_IN + lk;
    const float* __restrict__ b0p  = W + lm;        // N column lm
    const float* __restrict__ b1p  = W + 16 + lm;   // N column 16+lm

    v8f acc0 = {0.f,0.f,0.f,0.f,0.f,0.f,0.f,0.f};
    v8f acc1 = {0.f,0.f,0.f,0.f,0.f,0.f,0.f,0.f};

#pragma unroll
    for (int k = 0; k < F_IN; k += 4) {
        v2f a = *(const v2f*)(aptr + k);            // x[arow][k+lk .. k+lk+1]
        const int kb = k + lk;
        v2f b0, b1;
        b0.x = b0p[(size_t)kb * F_OUT];
        b0.y = b0p[(size_t)(kb + 1) * F_OUT];
        b1.x = b1p[(size_t)kb * F_OUT];
        b1.y = b1p[(size_t)(kb + 1) * F_OUT];
        acc0 = __builtin_amdgcn_wmma_f32_16x16x4_f32(false, a, false, b0,
                                                     (short)0, acc0, false, false);
        acc1 = __builtin_amdgcn_wmma_f32_16x16x4_f32(false, a, false, b1,
                                                     (short)0, acc1, false, false);
    }

    const int moff = (lane >> 4) << 3;              // 0 or 8
#pragma unroll
    for (int r = 0; r < 8; ++r) {
        int m = row0 + moff + r;
        if (m < N) {
            h[(size_t)m * F_OUT + lm]      = acc0[r];
            h[(size_t)m * F_OUT + 16 + lm] = acc1[r];
        }
    }
}

// ------------------------------------- out = b + h * dinv^2 (self loops) ---
__global__ void gcn_init_out(const float* __restrict__ h,
                             const float* __restrict__ dinv,
                             const float* __restrict__ b,
                             float* __restrict__ out, int N) {
    int i = blockIdx.x * blockDim.x + threadIdx.x;
    if (i < N * F_OUT) {
        int n = i >> 5;                             // F_OUT == 32
        int f = i & 31;
        float s = dinv[n];
        out[i] = b[f] + h[i] * s * s;
    }
}

// ------------------- out[dst] += h[src] * dinv[src]*dinv[dst] (per edge) ---
// One wave per edge, 32 lanes = 32 features: the h gather and the atomic
// burst are each a single coalesced 128B transaction; src/dst/dinv loads are
// wave-uniform. h and out are L2-resident (12.8 MB each, 192 MB L2).
__global__ void gcn_scatter(const int* __restrict__ src,
                            const int* __restrict__ dst,
                            const float* __restrict__ h,
                            const float* __restrict__ dinv,
                            float* __restrict__ out, int E) {
    int tid = blockIdx.x * blockDim.x + threadIdx.x;
    int e = tid >> 5;
    int f = tid & 31;
    if (e < E) {
        int s = src[e];
        int d = dst[e];
        float nrm = dinv[s] * dinv[d];
        atomicAdd(&out[(size_t)d * F_OUT + f], h[(size_t)s * F_OUT + f] * nrm);
    }
}

// ---------------------------------------------------------------- launch ----
extern "C" void kernel_launch(void* const* d_in, const int* in_sizes, int n_in,
                              void* d_out, int out_size, void* d_ws, size_t ws_size,
                              hipStream_t stream) {
    const float* x    = (const float*)d_in[0];   // [N, 128]
    const int*   edge = (const int*)  d_in[1];   // [2, E]
    const float* W    = (const float*)d_in[2];   // [128, 32]
    const float* b    = (const float*)d_in[3];   // [32]
    float*       out  = (float*)d_out;           // [N, 32]

    const int N = in_sizes[0] / F_IN;
    const int E = in_sizes[1] / 2;
    const int* src = edge;
    const int* dst = edge + E;

    // workspace: deg[N] | dinv[N] | h[N*32]  ->  34*N*4 bytes (~13.6 MB)
    float* deg  = (float*)d_ws;
    float* dinv = deg + N;
    float* h    = dinv + N;

    const int T = 256;

    gcn_init_deg <<<(N + T - 1) / T, T, 0, stream>>>(deg, N);
    gcn_count_deg<<<(E + T - 1) / T, T, 0, stream>>>(dst, deg, E);
    gcn_dinv     <<<(N + T - 1) / T, T, 0, stream>>>(deg, dinv, N);

    const int mtiles = (N + 15) / 16;            // 6250 for N=100000
    gcn_gemm_wmma<<<(mtiles + 7) / 8, T, 0, stream>>>(x, W, h, N);

    gcn_init_out <<<((size_t)N * F_OUT + T - 1) / T, T, 0, stream>>>(h, dinv, b, out, N);
    gcn_scatter  <<<((size_t)E * 32 + T - 1) / T, T, 0, stream>>>(src, dst, h, dinv, out, E);
}